// GAT_85332410237413
// MI455X (gfx1250) — compile-verified
//
#include <hip/hip_runtime.h>
#include <math.h>

typedef __attribute__((ext_vector_type(2))) float v2f;
typedef __attribute__((ext_vector_type(8))) float v8f;

#define NEG_INF_F (-9.0e15f)
#define LRELU_ALPHA 0.2f

// D = A(16x4,f32) * B(4x16,f32) + C(16x16,f32), wave32.
// 8-arg form: (neg_a, A, neg_b, B, c_mod, C, reuse_a, reuse_b)
static __device__ __forceinline__ v8f wmma_f32(v2f a, v2f b, v8f c) {
    return __builtin_amdgcn_wmma_f32_16x16x4_f32(false, a, false, b, (short)0, c, false, false);
}

static __device__ __forceinline__ float eluf(float v) {
    return v > 0.0f ? v : (__expf(v) - 1.0f);
}

// ---------------------------------------------------------------------------
// Kernel 1: conv1d(k=7,pad=3) + mean over L, collapsed to a sum + edge fixups.
// feat[r, c] = (S * sum_k w[c,k] - sum_k w[c,k]*edge_k) / L + conv_b[c]
// edge: k=0..2 -> tail_{3-k}, k=4..6 -> head_{k-3}
// ---------------------------------------------------------------------------
__global__ __launch_bounds__(256) void feat_kernel(
    const float* __restrict__ x,       // (768, 4096)
    const float* __restrict__ conv_w,  // (128, 7)
    const float* __restrict__ conv_b,  // (128)
    float* __restrict__ feat)          // (768, 128)
{
    __shared__ float red[256];
    __shared__ float bv[7]; // S, h1,h2,h3, t1,t2,t3
    const int r = blockIdx.x;
    const int t = threadIdx.x;

    const float4* xr4 = (const float4*)(x + (size_t)r * 4096);
    float s = 0.0f;
    for (int i = t; i < 1024; i += 256) {
        float4 v = xr4[i];
        s += v.x + v.y + v.z + v.w;
    }
    red[t] = s;
    __syncthreads();
    for (int st = 128; st > 0; st >>= 1) {
        if (t < st) red[t] += red[t + st];
        __syncthreads();
    }
    if (t == 0) {
        const float* xp = x + (size_t)r * 4096;
        bv[0] = red[0];
        bv[1] = xp[0];
        bv[2] = xp[0] + xp[1];
        bv[3] = xp[0] + xp[1] + xp[2];
        bv[4] = xp[4095];
        bv[5] = xp[4095] + xp[4094];
        bv[6] = xp[4095] + xp[4094] + xp[4093];
    }
    __syncthreads();
    if (t < 128) {
        const float* w = conv_w + t * 7;
        const float S = bv[0];
        const float wsum = w[0] + w[1] + w[2] + w[3] + w[4] + w[5] + w[6];
        const float corr = w[0] * bv[6] + w[1] * bv[5] + w[2] * bv[4]
                         + w[4] * bv[1] + w[5] * bv[2] + w[6] * bv[3];
        feat[(size_t)r * 128 + t] = (S * wsum - corr) * (1.0f / 4096.0f) + conv_b[t];
    }
}

// ---------------------------------------------------------------------------
// Kernel 2: one (batch, head) per block.
// Wh = feat[b](12x128, padded to 16) @ W_h(128x256) via WMMA f32 16x16x4.
// f1/f2, masked softmax -> att(16x16, zero-padded), h' = att @ Wh via WMMA,
// ELU, scatter into hcat[b, :, h*256 : (h+1)*256].
// ---------------------------------------------------------------------------
__global__ __launch_bounds__(256) void gat_head_kernel(
    const float* __restrict__ feat,     // (768, 128)
    const float* __restrict__ adj,      // (64, 12, 12)
    const float* __restrict__ W_heads,  // (8, 128, 256)
    const float* __restrict__ a_heads,  // (8, 512)
    float* __restrict__ hcat)           // (64, 12, 2048)
{
    __shared__ float As[16][128];   // feat tile, rows 12..15 zero
    __shared__ float Wh[16][256];
    __shared__ float att[16][16];
    __shared__ float a1s[256], a2s[256];
    __shared__ float f1s[16], f2s[16];
    __shared__ float red1[256], red2[256];

    const int b = blockIdx.x;
    const int h = blockIdx.y;
    const int t = threadIdx.x;
    const int lane = t & 31;
    const int wave = t >> 5;        // 8 waves
    const int ll = lane & 15;       // lane-low: column / M index
    const int lh = lane >> 4;       // lane-half

    for (int idx = t; idx < 16 * 128; idx += 256) {
        int i = idx >> 7, k = idx & 127;
        As[i][k] = (i < 12) ? feat[((size_t)b * 12 + i) * 128 + k] : 0.0f;
    }
    a1s[t] = a_heads[h * 512 + t];
    a2s[t] = a_heads[h * 512 + 256 + t];
    __syncthreads();

    // ---- GEMM1: Wh = As @ W_h; wave handles N-tiles 2w, 2w+1 ----
    const float* Wg = W_heads + (size_t)h * 128 * 256;
    const int c0 = wave * 32 + ll;
    v8f acc0 = {0.f,0.f,0.f,0.f,0.f,0.f,0.f,0.f};
    v8f acc1 = {0.f,0.f,0.f,0.f,0.f,0.f,0.f,0.f};
    for (int k4 = 0; k4 < 128; k4 += 4) {
        const int kA = k4 + (lh << 1);
        v2f a; a.x = As[ll][kA]; a.y = As[ll][kA + 1];
        const float* bp = Wg + (size_t)kA * 256 + c0;
        v2f b0; b0.x = bp[0];  b0.y = bp[256];
        v2f b1; b1.x = bp[16]; b1.y = bp[256 + 16];
        acc0 = wmma_f32(a, b0, acc0);
        acc1 = wmma_f32(a, b1, acc1);
    }
    // D layout: reg r -> M = r + 8*lane_half, N = lane&15
    for (int r = 0; r < 8; ++r) {
        const int m = r + (lh << 3);
        Wh[m][wave * 32 + ll]      = acc0[r];
        Wh[m][wave * 32 + 16 + ll] = acc1[r];
    }
    __syncthreads();

    // ---- f1 = Wh @ a1, f2 = Wh @ a2 (16 partials per row) ----
    {
        const int i = t >> 4, part = t & 15;
        float p1 = 0.0f, p2 = 0.0f;
        for (int o = part; o < 256; o += 16) {
            const float v = Wh[i][o];
            p1 += v * a1s[o];
            p2 += v * a2s[o];
        }
        red1[t] = p1; red2[t] = p2;
    }
    __syncthreads();
    if ((t & 15) == 0) {
        const int i = t >> 4;
        float s1 = 0.0f, s2 = 0.0f;
        for (int q = 0; q < 16; ++q) { s1 += red1[(i << 4) + q]; s2 += red2[(i << 4) + q]; }
        f1s[i] = s1; f2s[i] = s2;
    }
    __syncthreads();

    // ---- masked softmax over j, att padded with zeros ----
    if (t < 16) {
        const int i = t;
        if (i < 12) {
            const float* ar = adj + ((size_t)b * 12 + i) * 12;
            float ev[12];
            float mx = -3.4e38f;
            for (int j = 0; j < 12; ++j) {
                float e = f1s[i] + f2s[j];
                e = (e >= 0.0f) ? e : LRELU_ALPHA * e;
                e = (ar[j] > 0.0f) ? e : NEG_INF_F;
                ev[j] = e;
                mx = fmaxf(mx, e);
            }
            float ssum = 0.0f;
            for (int j = 0; j < 12; ++j) { ev[j] = __expf(ev[j] - mx); ssum += ev[j]; }
            const float inv = 1.0f / ssum;
            for (int j = 0; j < 12; ++j) att[i][j] = ev[j] * inv;
            att[i][12] = att[i][13] = att[i][14] = att[i][15] = 0.0f;
        } else {
            for (int j = 0; j < 16; ++j) att[i][j] = 0.0f;
        }
    }
    __syncthreads();

    // ---- GEMM2: h' = att(16x16) @ Wh(16x256), then ELU + store ----
    v8f d0 = {0.f,0.f,0.f,0.f,0.f,0.f,0.f,0.f};
    v8f d1 = {0.f,0.f,0.f,0.f,0.f,0.f,0.f,0.f};
    for (int k4 = 0; k4 < 16; k4 += 4) {
        const int kA = k4 + (lh << 1);
        v2f a; a.x = att[ll][kA]; a.y = att[ll][kA + 1];
        v2f b0; b0.x = Wh[kA][wave * 32 + ll];      b0.y = Wh[kA + 1][wave * 32 + ll];
        v2f b1; b1.x = Wh[kA][wave * 32 + 16 + ll]; b1.y = Wh[kA + 1][wave * 32 + 16 + ll];
        d0 = wmma_f32(a, b0, d0);
        d1 = wmma_f32(a, b1, d1);
    }
    for (int r = 0; r < 8; ++r) {
        const int m = r + (lh << 3);
        if (m < 12) {
            const size_t base = ((size_t)b * 12 + m) * 2048 + h * 256;
            hcat[base + wave * 32 + ll]      = eluf(d0[r]);
            hcat[base + wave * 32 + 16 + ll] = eluf(d1[r]);
        }
    }
}

// ---------------------------------------------------------------------------
// Kernel 3: output GAT layer + fused FC. One batch per block.
// Wh2 = hcat[b](12x2048) @ out_W(2048x256) with K chunked into 512-wide LDS
// tiles; attention; h' = att @ Wh2 via WMMA; ELU; logits = out @ fc_w^T + fc_b.
// ---------------------------------------------------------------------------
__global__ __launch_bounds__(256) void gat_out_kernel(
    const float* __restrict__ hcat,   // (64, 12, 2048)
    const float* __restrict__ adj,    // (64, 12, 12)
    const float* __restrict__ out_W,  // (2048, 256)
    const float* __restrict__ out_a,  // (512)
    const float* __restrict__ fc_w,   // (12, 3072)
    const float* __restrict__ fc_b,   // (12)
    float* __restrict__ logits)       // (64, 12)
{
    __shared__ float As[16 * 512];    // A chunk; later reused as out(16x256) flat
    __shared__ float Wh[16][256];
    __shared__ float att[16][16];
    __shared__ float a1s[256], a2s[256];
    __shared__ float f1s[16], f2s[16];
    __shared__ float red1[256], red2[256];

    const int b = blockIdx.x;
    const int t = threadIdx.x;
    const int lane = t & 31;
    const int wave = t >> 5;
    const int ll = lane & 15;
    const int lh = lane >> 4;

    a1s[t] = out_a[t];
    a2s[t] = out_a[256 + t];

    // ---- GEMM1: K=2048 in 4 chunks of 512 ----
    const int c0 = wave * 32 + ll;
    v8f acc0 = {0.f,0.f,0.f,0.f,0.f,0.f,0.f,0.f};
    v8f acc1 = {0.f,0.f,0.f,0.f,0.f,0.f,0.f,0.f};
    for (int kc = 0; kc < 2048; kc += 512) {
        for (int idx = t; idx < 16 * 512; idx += 256) {
            const int i = idx >> 9, k = idx & 511;
            As[i * 512 + k] = (i < 12) ? hcat[((size_t)b * 12 + i) * 2048 + kc + k] : 0.0f;
        }
        __syncthreads();
        for (int k4 = 0; k4 < 512; k4 += 4) {
            const int kA = k4 + (lh << 1);
            v2f a; a.x = As[ll * 512 + kA]; a.y = As[ll * 512 + kA + 1];
            const float* bp = out_W + (size_t)(kc + kA) * 256 + c0;
            v2f b0; b0.x = bp[0];  b0.y = bp[256];
            v2f b1; b1.x = bp[16]; b1.y = bp[256 + 16];
            acc0 = wmma_f32(a, b0, acc0);
            acc1 = wmma_f32(a, b1, acc1);
        }
        __syncthreads();
    }
    for (int r = 0; r < 8; ++r) {
        const int m = r + (lh << 3);
        Wh[m][wave * 32 + ll]      = acc0[r];
        Wh[m][wave * 32 + 16 + ll] = acc1[r];
    }
    __syncthreads();

    // ---- f1/f2 ----
    {
        const int i = t >> 4, part = t & 15;
        float p1 = 0.0f, p2 = 0.0f;
        for (int o = part; o < 256; o += 16) {
            const float v = Wh[i][o];
            p1 += v * a1s[o];
            p2 += v * a2s[o];
        }
        red1[t] = p1; red2[t] = p2;
    }
    __syncthreads();
    if ((t & 15) == 0) {
        const int i = t >> 4;
        float s1 = 0.0f, s2 = 0.0f;
        for (int q = 0; q < 16; ++q) { s1 += red1[(i << 4) + q]; s2 += red2[(i << 4) + q]; }
        f1s[i] = s1; f2s[i] = s2;
    }
    __syncthreads();

    // ---- masked softmax ----
    if (t < 16) {
        const int i = t;
        if (i < 12) {
            const float* ar = adj + ((size_t)b * 12 + i) * 12;
            float ev[12];
            float mx = -3.4e38f;
            for (int j = 0; j < 12; ++j) {
                float e = f1s[i] + f2s[j];
                e = (e >= 0.0f) ? e : LRELU_ALPHA * e;
                e = (ar[j] > 0.0f) ? e : NEG_INF_F;
                ev[j] = e;
                mx = fmaxf(mx, e);
            }
            float ssum = 0.0f;
            for (int j = 0; j < 12; ++j) { ev[j] = __expf(ev[j] - mx); ssum += ev[j]; }
            const float inv = 1.0f / ssum;
            for (int j = 0; j < 12; ++j) att[i][j] = ev[j] * inv;
            att[i][12] = att[i][13] = att[i][14] = att[i][15] = 0.0f;
        } else {
            for (int j = 0; j < 16; ++j) att[i][j] = 0.0f;
        }
    }
    __syncthreads();

    // ---- GEMM2: h' = att @ Wh2; ELU; store flat into As (reused) ----
    v8f d0 = {0.f,0.f,0.f,0.f,0.f,0.f,0.f,0.f};
    v8f d1 = {0.f,0.f,0.f,0.f,0.f,0.f,0.f,0.f};
    for (int k4 = 0; k4 < 16; k4 += 4) {
        const int kA = k4 + (lh << 1);
        v2f a; a.x = att[ll][kA]; a.y = att[ll][kA + 1];
        v2f b0; b0.x = Wh[kA][wave * 32 + ll];      b0.y = Wh[kA + 1][wave * 32 + ll];
        v2f b1; b1.x = Wh[kA][wave * 32 + 16 + ll]; b1.y = Wh[kA + 1][wave * 32 + 16 + ll];
        d0 = wmma_f32(a, b0, d0);
        d1 = wmma_f32(a, b1, d1);
    }
    for (int r = 0; r < 8; ++r) {
        const int m = r + (lh << 3);
        As[m * 256 + wave * 32 + ll]      = eluf(d0[r]);
        As[m * 256 + wave * 32 + 16 + ll] = eluf(d1[r]);
    }
    __syncthreads();

    // ---- fused FC: logits[b, c] = out_flat . fc_w[c] + fc_b[c] ----
    for (int c = 0; c < 12; ++c) {
        float p = 0.0f;
        for (int f = t; f < 3072; f += 256) p += As[f] * fc_w[c * 3072 + f];
        red1[t] = p;
        __syncthreads();
        for (int st = 128; st > 0; st >>= 1) {
            if (t < st) red1[t] += red1[t + st];
            __syncthreads();
        }
        if (t == 0) logits[b * 12 + c] = red1[0] + fc_b[c];
        __syncthreads();
    }
}

// ---------------------------------------------------------------------------
extern "C" void kernel_launch(void* const* d_in, const int* in_sizes, int n_in,
                              void* d_out, int out_size, void* d_ws, size_t ws_size,
                              hipStream_t stream) {
    const float* x       = (const float*)d_in[0];  // (64,12,1,4096)
    const float* adj     = (const float*)d_in[1];  // (64,12,12)
    const float* conv_w  = (const float*)d_in[2];  // (128,1,7)
    const float* conv_b  = (const float*)d_in[3];  // (128)
    const float* W_heads = (const float*)d_in[4];  // (8,128,256)
    const float* a_heads = (const float*)d_in[5];  // (8,512,1)
    const float* out_W   = (const float*)d_in[6];  // (2048,256)
    const float* out_a   = (const float*)d_in[7];  // (512,1)
    const float* fc_w    = (const float*)d_in[8];  // (12,3072)
    const float* fc_b    = (const float*)d_in[9];  // (12)
    float* out = (float*)d_out;                    // (64,12)

    float* feat = (float*)d_ws;                    // 768*128 floats
    float* hcat = feat + 768 * 128;                // 64*12*2048 floats

    feat_kernel<<<768, 256, 0, stream>>>(x, conv_w, conv_b, feat);
    gat_head_kernel<<<dim3(64, 8), 256, 0, stream>>>(feat, adj, W_heads, a_heads, hcat);
    gat_out_kernel<<<64, 256, 0, stream>>>(hcat, adj, out_W, out_a, fc_w, fc_b, out);
}